// auxiliary_loss_23819888623694
// MI455X (gfx1250) — compile-verified
//
#include <hip/hip_runtime.h>

// CDNA5 / gfx1250: wave32, WMMA 16x16x4 f32 used for cross-lane reduction.

typedef float v2f __attribute__((ext_vector_type(2)));
typedef float v8f __attribute__((ext_vector_type(8)));
typedef float vf4 __attribute__((ext_vector_type(4)));
typedef int   vi4 __attribute__((ext_vector_type(4)));

// Full 32-lane sum using V_WMMA_F32_16X16X4_F32.
// A layout (documented): lane L, VGPR0 -> (M = L%16, K = 0 or 2); VGPR1 -> K = 1 or 3.
// We place the per-lane partial in A.vgpr0 and 0 in A.vgpr1; B = all ones
// (value-uniform, so its undocumented lane layout is irrelevant).
// Then D[m][n] = v_m + v_{m+16}.  D vgpr j, lanes 0-15 hold row j, lanes 16-31 row j+8.
// Sum of the 8 D vgprs gives half-total per lane group; one shfl_xor(16) completes it.
__device__ __forceinline__ float wave_reduce_f32(float v) {
    v2f a; a.x = v;    a.y = 0.0f;
    v2f b; b.x = 1.0f; b.y = 1.0f;
    v8f c = {};
    v8f d = __builtin_amdgcn_wmma_f32_16x16x4_f32(
        /*neg_a=*/false, a, /*neg_b=*/false, b,
        /*c_mod=*/(short)0, c, /*reuse_a=*/false, /*reuse_b=*/false);
    float s = ((d[0] + d[1]) + (d[2] + d[3])) + ((d[4] + d[5]) + (d[6] + d[7]));
    s += __shfl_xor(s, 16, 32);
    return s;
}

__global__ void __launch_bounds__(256)
auxloss_reduce_kernel(const float* __restrict__ y,
                      const int*  __restrict__ lab,
                      float* __restrict__ partials,
                      int n4, int n) {
    const vf4* __restrict__ y4 = (const vf4*)y;
    const vi4* __restrict__ l4 = (const vi4*)lab;

    float s_all = 0.0f;
    float s_pos = 0.0f;
    int   cnt   = 0;

    const int tid    = blockIdx.x * blockDim.x + threadIdx.x;
    const int stride = gridDim.x * blockDim.x;

    // Main streaming loop: 16B+16B non-temporal loads per iteration (read-once data).
    for (int i = tid; i < n4; i += stride) {
        vf4 yv = __builtin_nontemporal_load(&y4[i]);
        vi4 lv = __builtin_nontemporal_load(&l4[i]);
        s_all += (yv.x + yv.y) + (yv.z + yv.w);
        s_pos += (lv.x ? yv.x : 0.0f) + (lv.y ? yv.y : 0.0f)
               + (lv.z ? yv.z : 0.0f) + (lv.w ? yv.w : 0.0f);
        cnt   += (lv.x + lv.y) + (lv.z + lv.w);
    }
    // Scalar tail (n not divisible by 4) — robustness; empty for N = 2^25.
    for (int i = n4 * 4 + tid; i < n; i += stride) {
        float yv = __builtin_nontemporal_load(&y[i]);
        int   lv = __builtin_nontemporal_load(&lab[i]);
        s_all += yv;
        s_pos += lv ? yv : 0.0f;
        cnt   += lv;
    }

    // All 256 threads reach here (EXEC all ones, as WMMA requires).
    float w_all = wave_reduce_f32(s_all);
    float w_pos = wave_reduce_f32(s_pos);
    float w_cnt = wave_reduce_f32((float)cnt);   // <= 4096 per wave: exact in f32

    __shared__ float sh[3][8];
    const int wave = threadIdx.x >> 5;
    const int lane = threadIdx.x & 31;
    if (lane == 0) {
        sh[0][wave] = w_all;
        sh[1][wave] = w_pos;
        sh[2][wave] = w_cnt;
    }
    __syncthreads();
    if (threadIdx.x == 0) {
        float a = 0.0f, p = 0.0f, q = 0.0f;
        #pragma unroll
        for (int j = 0; j < 8; ++j) { a += sh[0][j]; p += sh[1][j]; q += sh[2][j]; }
        partials[3 * blockIdx.x + 0] = a;
        partials[3 * blockIdx.x + 1] = p;
        partials[3 * blockIdx.x + 2] = q;
    }
}

__global__ void __launch_bounds__(256)
auxloss_finalize_kernel(const float* __restrict__ partials,
                        float* __restrict__ out,
                        int nblocks, float n_total) {
    __shared__ double sh[256][3];
    double a = 0.0, p = 0.0, q = 0.0;
    for (int i = threadIdx.x; i < nblocks; i += 256) {
        a += (double)partials[3 * i + 0];
        p += (double)partials[3 * i + 1];
        q += (double)partials[3 * i + 2];
    }
    sh[threadIdx.x][0] = a;
    sh[threadIdx.x][1] = p;
    sh[threadIdx.x][2] = q;
    __syncthreads();
    if (threadIdx.x == 0) {
        double A = 0.0, P = 0.0, C = 0.0;
        for (int j = 0; j < 256; ++j) { A += sh[j][0]; P += sh[j][1]; C += sh[j][2]; }
        double pos     = C;
        double neg     = (double)n_total - C;
        double sum_pos = P;
        double sum_neg = A - P;
        bool   valid   = (pos > 0.0) && (neg > 0.0);
        double denom   = valid ? (pos * neg) : 1.0;
        double loss    = 0.2 * (pos * sum_neg - neg * sum_pos) / denom;
        out[0] = valid ? (float)loss : 0.0f;
    }
}

extern "C" void kernel_launch(void* const* d_in, const int* in_sizes, int n_in,
                              void* d_out, int out_size, void* d_ws, size_t ws_size,
                              hipStream_t stream) {
    const float* y   = (const float*)d_in[0];
    const int*   lab = (const int*)d_in[1];
    float*       ws  = (float*)d_ws;
    float*       out = (float*)d_out;

    const int n  = in_sizes[0];
    const int n4 = n >> 2;

    int blocks = 1024;                                   // 12 KB of partials in d_ws
    size_t need = (size_t)blocks * 3 * sizeof(float);
    if (ws_size < need) {
        blocks = (int)(ws_size / (3 * sizeof(float)));
        if (blocks < 1) blocks = 1;
    }

    auxloss_reduce_kernel<<<blocks, 256, 0, stream>>>(y, lab, ws, n4, n);
    auxloss_finalize_kernel<<<1, 256, 0, stream>>>(ws, out, blocks, (float)n);
}